// cheb_conv_withSAt_21414706938396
// MI455X (gfx1250) — compile-verified
//
#include <hip/hip_runtime.h>

#define B_DIM  16
#define N_DIM  1536
#define FIN    64
#define T_DIM  12
#define K_CHEB 3
#define FOUT   64
#define CDIM   (FIN * T_DIM)   // 768 fused (f,t) columns, t fastest (matches x layout)

typedef float v2f __attribute__((ext_vector_type(2)));
typedef float v4f __attribute__((ext_vector_type(4)));
typedef float v8f __attribute__((ext_vector_type(8)));

// LDS strides (floats):
//  PST: x-chunk pair layout [n/2][c][2]; 1568 % 64 == 32 -> the two half-waves
//       of a b64 B-fragment load land in disjoint 32-bank halves (conflict-free).
//  AST: A-tile [m][n]; 20*m mod 64 distinct multiples of 4 -> disjoint 4-bank
//       footprints for each lane-pair's b64 A-fragment load.
//  RST: rhs pair layout [m][f/2][t][2]; 772 % 64 == 4 -> distinct banks over m.
#define PST 1568
#define AST 20
#define RST 772

__global__ __launch_bounds__(256)
void cheb_conv_sat_kernel(const float* __restrict__ x,     // (B,N,FIN,T)
                          const float* __restrict__ At,    // (B,N,N)
                          const float* __restrict__ cheb,  // (K,N,N)
                          const float* __restrict__ Theta, // (K,FIN,FOUT)
                          float* __restrict__ out) {       // (B,N,FOUT,T)
  // Union buffer: x-chunk pair layout (8*1568 = 12544 floats) aliased as the
  // rhs pair layout (max index 12347) between the two GEMM phases.
  __shared__ __align__(16) float ubuf[8 * PST];
  __shared__ __align__(8)  float abuf[16 * AST];

  const int tid  = threadIdx.x;
  const int lane = tid & 31;
  const int wave = tid >> 5;
  const int half = lane >> 4;   // 0: lanes 0-15, 1: lanes 16-31
  const int l16  = lane & 15;

  const int m0 = blockIdx.x * 16;
  const int b  = blockIdx.y;

  const float* xb  = x  + (size_t)b * N_DIM * CDIM;
  const float* Atb = At + (size_t)b * N_DIM * N_DIM;

  // Second-GEMM ownership: wave -> o-group (16 outputs) and 6 timesteps.
  const int og    = wave >> 1;
  const int tbase = (wave & 1) * 6;

  v8f outacc[6];
  for (int i = 0; i < 6; ++i)
    for (int e = 0; e < 8; ++e) outacc[i][e] = 0.0f;

  for (int k = 0; k < K_CHEB; ++k) {
    const float* chk = cheb  + (size_t)k * N_DIM * N_DIM;
    const float* thk = Theta + (size_t)k * FIN * FOUT;

    v8f rhsacc[6];
    for (int i = 0; i < 6; ++i)
      for (int e = 0; e < 8; ++e) rhsacc[i][e] = 0.0f;

    // ---------- GEMM 1: rhs[m, c] = sum_n (cheb[k]*At[b])[n, m] * x[b][n, c]
    for (int nc = 0; nc < N_DIM / 16; ++nc) {
      const int n0 = nc * 16;
      __syncthreads();  // previous chunk (or previous k's rhs reads) done

      // Stage A tile transposed [m][n], forming cheb*At on the fly.
      {
        const int an = tid >> 4, am = tid & 15;
        const size_t g = (size_t)(n0 + an) * N_DIM + (size_t)(m0 + am);
        abuf[am * AST + an] = chk[g] * Atb[g];
      }
      // Stage X tile (16 n x 768 c) into the n-pair-interleaved layout:
      // each thread handles 6 blocks of (2 n x 4 c): two row v4f loads,
      // shuffled into two interleaved b128 stores.
      for (int r = 0; r < 6; ++r) {
        const int g  = r * 256 + tid;        // block index in [0, 1536)
        const int np = g / 192;              // 192 (4c)-blocks per n-pair
        const int cq = (g % 192) * 4;
        const float* s0 = xb + (size_t)(n0 + 2 * np) * CDIM + cq;
        const v4f a0 = *(const v4f*)s0;
        const v4f a1 = *(const v4f*)(s0 + CDIM);
        if (nc + 1 < N_DIM / 16)
          __builtin_prefetch(s0 + 16 * CDIM, 0, 3);  // global_prefetch_b8
        v4f w0, w1;
        w0[0] = a0[0]; w0[1] = a1[0]; w0[2] = a0[1]; w0[3] = a1[1];
        w1[0] = a0[2]; w1[1] = a1[2]; w1[2] = a0[3]; w1[3] = a1[3];
        float* d = &ubuf[np * PST + cq * 2];
        *(v4f*)d       = w0;
        *(v4f*)(d + 4) = w1;
      }
      __syncthreads();

      for (int j = 0; j < 4; ++j) {
        // A fragment: M = l16, K rows {4j+2h, 4j+2h+1} -> one ds_load_b64
        const v2f afrag = *(const v2f*)&abuf[l16 * AST + 4 * j + 2 * half];
        for (int i = 0; i < 6; ++i) {
          const int c = (wave * 6 + i) * 16 + l16;
          // B fragment: K rows {4j+2h, 4j+2h+1} at column c -> one ds_load_b64
          const v2f bfrag = *(const v2f*)&ubuf[(2 * j + half) * PST + 2 * c];
          rhsacc[i] = __builtin_amdgcn_wmma_f32_16x16x4_f32(
              false, afrag, false, bfrag, (short)0, rhsacc[i], false, false);
        }
      }
    }

    // ---------- rhs -> LDS in f-pair-interleaved layout [m][f/2][t][2]
    __syncthreads();
    for (int i = 0; i < 6; ++i) {
      const int c = (wave * 6 + i) * 16 + l16;
      const int f = c / T_DIM;
      const int t = c % T_DIM;
      const int off = (f >> 1) * (2 * T_DIM) + 2 * t + (f & 1);
      for (int g = 0; g < 8; ++g) {
        const int m = g + half * 8;     // D layout: VGPR g -> M = g (+8 hi half)
        ubuf[m * RST + off] = rhsacc[i][g];
      }
    }
    __syncthreads();

    // ---------- GEMM 2: out[m, o, t] += sum_f rhs[m, f*12+t] * Theta[k][f, o]
    for (int j = 0; j < 16; ++j) {
      v2f bfrag;  // Theta fragment from global (16 KB, L1/L2 resident)
      {
        const int o = og * 16 + l16;
        bfrag[0] = thk[(4 * j + half * 2 + 0) * FOUT + o];
        bfrag[1] = thk[(4 * j + half * 2 + 1) * FOUT + o];
      }
      for (int i = 0; i < 6; ++i) {
        const int t = tbase + i;
        // rhs fragment: M = l16, K rows {4j+2h, 4j+2h+1} -> one ds_load_b64
        const v2f afrag =
            *(const v2f*)&ubuf[l16 * RST + (2 * j + half) * (2 * T_DIM) + 2 * t];
        outacc[i] = __builtin_amdgcn_wmma_f32_16x16x4_f32(
            false, afrag, false, bfrag, (short)0, outacc[i], false, false);
      }
    }
    __syncthreads();  // ubuf reused by next k's staging
  }

  // ---------- ReLU + store: out[b][m0+M][o][t]
  float* ob = out + (size_t)b * N_DIM * FOUT * T_DIM;
  for (int i = 0; i < 6; ++i) {
    const int t = tbase + i;
    const int o = og * 16 + l16;
    for (int g = 0; g < 8; ++g) {
      const int m = m0 + g + half * 8;
      const float v = outacc[i][g];
      ob[((size_t)m * FOUT + o) * T_DIM + t] = v > 0.0f ? v : 0.0f;
    }
  }
}

extern "C" void kernel_launch(void* const* d_in, const int* in_sizes, int n_in,
                              void* d_out, int out_size, void* d_ws, size_t ws_size,
                              hipStream_t stream) {
  const float* x     = (const float*)d_in[0];  // (B,N,FIN,T)
  const float* At    = (const float*)d_in[1];  // (B,N,N)
  const float* cheb  = (const float*)d_in[2];  // (K,N,N)
  const float* Theta = (const float*)d_in[3];  // (K,FIN,FOUT)
  float* out = (float*)d_out;

  dim3 grid(N_DIM / 16, B_DIM);
  dim3 block(256);
  cheb_conv_sat_kernel<<<grid, block, 0, stream>>>(x, At, cheb, Theta, out);
}